// UMKD_48988396978318
// MI455X (gfx1250) — compile-verified
//
#include <hip/hip_runtime.h>
#include <hip/hip_bf16.h>
#include <math.h>

typedef __attribute__((ext_vector_type(2))) float v2f;
typedef __attribute__((ext_vector_type(8))) float v8f;

#define BATCH 128
#define NPTS  2048
#define CATE  55

// ---------------------------------------------------------------------------
// Kernel 1: per-batch argmax of cls_score (-> cid in workspace) + passthrough
// copy of cls_score to the output tail.
// ---------------------------------------------------------------------------
__global__ void argmax_copy_kernel(const float* __restrict__ cls,
                                   int* __restrict__ cid,
                                   float* __restrict__ out_cls) {
    const int b = blockIdx.x;      // 128 blocks
    const int t = threadIdx.x;     // 64 threads
    __shared__ float s[64];
    float val = -__builtin_inff();
    if (t < CATE) {
        val = cls[b * CATE + t];
        out_cls[b * CATE + t] = val;   // passthrough output
    }
    s[t] = val;
    __syncthreads();
    if (t == 0) {
        int best = 0;
        float bv = s[0];
        for (int i = 1; i < CATE; ++i) {
            if (s[i] > bv) { bv = s[i]; best = i; }   // first-max, like argmax
        }
        cid[b] = best;
    }
}

// ---------------------------------------------------------------------------
// Kernel 2: z = feat + relu(W[cid] @ feat + b[cid])  via V_WMMA_F32_16X16X4_F32
//
// One wave32 owns one (batch, 16-wide n-tile) column strip and keeps all
// K/16 m-tile accumulators in registers; the B fragment (feat) is loaded once
// per k-step and reused across every m-tile's WMMA.
//
// f32 WMMA fragment layouts (ISA 7.12.2):
//   A 16x4 : lane<16 -> (M=lane,   K=k+0 / k+1 in v0/v1)
//            lane>=16-> (M=lane-16,K=k+2 / k+3 in v0/v1)
//   B 4x16 : lane<16 -> (K=k+0/k+1, N=lane),  lane>=16 -> (K=k+2/k+3, N=lane-16)
//   C/D    : VGPR r -> M = r + (lane>=16 ? 8 : 0), N = lane & 15
// ---------------------------------------------------------------------------
template <int K>
__global__ void gemm_off_kernel(const float* __restrict__ feat,  // [B,K,N]
                                const float* __restrict__ W,     // [C,K,K]
                                const float* __restrict__ bias,  // [C,K]
                                const int* __restrict__ cid,     // [B]
                                float* __restrict__ z) {         // [B,K,N]
    constexpr int N  = NPTS;
    constexpr int MT = K / 16;            // m-tiles held per wave

    const int lane   = threadIdx.x & 31;
    const int wave   = threadIdx.x >> 5;
    const int ntiles = N / 16;            // 128
    const int wg     = blockIdx.x * (blockDim.x >> 5) + wave;
    const int b      = wg / ntiles;       // wave-uniform
    const int nt     = wg % ntiles;
    if (b >= BATCH) return;               // wave-uniform guard (EXEC stays all-1s)

    const int c = cid[b];
    const float* __restrict__ Wc = W    + (size_t)c * K * K;
    const float* __restrict__ fb = feat + (size_t)b * K * N;

    const int n   = nt * 16 + (lane & 15);
    const int kk  = (lane >> 4) << 1;     // 0 for lanes 0-15, 2 for lanes 16-31
    const int row = lane & 15;

    v8f acc[MT] = {};

    for (int k = 0; k < K; k += 4) {
        v2f bmat;
        bmat.x = fb[(size_t)(k + kk)     * N + n];
        bmat.y = fb[(size_t)(k + kk + 1) * N + n];
#pragma unroll
        for (int mt = 0; mt < MT; ++mt) {
            const float* wrow = Wc + (size_t)(mt * 16 + row) * K + (k + kk);
            v2f amat;
            amat.x = wrow[0];
            amat.y = wrow[1];
            acc[mt] = __builtin_amdgcn_wmma_f32_16x16x4_f32(
                /*neg_a=*/false, amat, /*neg_b=*/false, bmat,
                /*c_mod=*/(short)0, acc[mt],
                /*reuse_a=*/false, /*reuse_b=*/false);
        }
    }

    // Fused epilogue: bias + relu + add feat, write z (softmax input) in place
    const int madd = (lane >> 4) * 8;     // 0 or 8
    float* __restrict__ zb = z + (size_t)b * K * N;
#pragma unroll
    for (int mt = 0; mt < MT; ++mt) {
#pragma unroll
        for (int r = 0; r < 8; ++r) {
            const int m  = mt * 16 + r + madd;
            float off    = acc[mt][r] + bias[(size_t)c * K + m];
            off          = off > 0.f ? off : 0.f;
            zb[(size_t)m * N + n] = fb[(size_t)m * N + n] + off;
        }
    }
}

// ---------------------------------------------------------------------------
// Kernel 3: in-place numerically-stable softmax over each row of 2048.
// One 256-thread block per (b,k) row; 8 elements per thread.
// ---------------------------------------------------------------------------
__global__ void softmax_rows_kernel(float* __restrict__ z) {
    constexpr int N = NPTS;
    float* __restrict__ p = z + (size_t)blockIdx.x * N;
    const int t = threadIdx.x;            // 256 threads

    float v[8];
    float m = -__builtin_inff();
#pragma unroll
    for (int i = 0; i < 8; ++i) {
        v[i] = p[t + i * 256];
        m = fmaxf(m, v[i]);
    }

    __shared__ float red[256];
    red[t] = m;
    __syncthreads();
    for (int s = 128; s > 0; s >>= 1) {
        if (t < s) red[t] = fmaxf(red[t], red[t + s]);
        __syncthreads();
    }
    m = red[0];
    __syncthreads();

    float sum = 0.f;
#pragma unroll
    for (int i = 0; i < 8; ++i) {
        v[i] = __expf(v[i] - m);
        sum += v[i];
    }
    red[t] = sum;
    __syncthreads();
    for (int s = 128; s > 0; s >>= 1) {
        if (t < s) red[t] += red[t + s];
        __syncthreads();
    }
    const float inv = 1.f / red[0];
#pragma unroll
    for (int i = 0; i < 8; ++i) p[t + i * 256] = v[i] * inv;
}

// ---------------------------------------------------------------------------
// Launch
// Inputs (setup_inputs order):
//  0 feat1[128,32,2048] 1 feat2[128,64,2048] 2 feat3[128,128,2048]
//  3 cls_score[128,55]  4 W1[55,32,32] 5 b1[55,32]
//  6 W2[55,64,64] 7 b2[55,64] 8 W3[55,128,128] 9 b3[55,128]
// Output: key_feat1 | key_feat2 | key_feat3 | cls_score (flat, f32)
// ---------------------------------------------------------------------------
extern "C" void kernel_launch(void* const* d_in, const int* in_sizes, int n_in,
                              void* d_out, int out_size, void* d_ws, size_t ws_size,
                              hipStream_t stream) {
    const float* feat1 = (const float*)d_in[0];
    const float* feat2 = (const float*)d_in[1];
    const float* feat3 = (const float*)d_in[2];
    const float* cls   = (const float*)d_in[3];
    const float* W1    = (const float*)d_in[4];
    const float* b1    = (const float*)d_in[5];
    const float* W2    = (const float*)d_in[6];
    const float* b2    = (const float*)d_in[7];
    const float* W3    = (const float*)d_in[8];
    const float* b3    = (const float*)d_in[9];

    float* out = (float*)d_out;
    const size_t o1 = 0;
    const size_t o2 = o1 + (size_t)BATCH * 32  * NPTS;
    const size_t o3 = o2 + (size_t)BATCH * 64  * NPTS;
    const size_t oc = o3 + (size_t)BATCH * 128 * NPTS;

    int* cid = (int*)d_ws;   // 128 ints of scratch

    // 1) argmax + cls passthrough
    argmax_copy_kernel<<<BATCH, 64, 0, stream>>>(cls, cid, out + oc);

    // 2) fused expert GEMM + bias + relu + residual  (z into d_out regions)
    // waves = B * (N/16) = 16384 per level; 8 waves/block -> 2048 blocks
    const int gemm_blocks = (BATCH * (NPTS / 16)) / 8;
    gemm_off_kernel<32> <<<gemm_blocks, 256, 0, stream>>>(feat1, W1, b1, cid, out + o1);
    gemm_off_kernel<64> <<<gemm_blocks, 256, 0, stream>>>(feat2, W2, b2, cid, out + o2);
    gemm_off_kernel<128><<<gemm_blocks, 256, 0, stream>>>(feat3, W3, b3, cid, out + o3);

    // 3) in-place row softmax over N=2048
    softmax_rows_kernel<<<BATCH * 32,  256, 0, stream>>>(out + o1);
    softmax_rows_kernel<<<BATCH * 64,  256, 0, stream>>>(out + o2);
    softmax_rows_kernel<<<BATCH * 128, 256, 0, stream>>>(out + o3);
}